// DenseAtt_39771397161679
// MI455X (gfx1250) — compile-verified
//
#include <hip/hip_runtime.h>

typedef float v2f __attribute__((ext_vector_type(2)));
typedef float v4f __attribute__((ext_vector_type(4)));
typedef float v8f __attribute__((ext_vector_type(8)));

#define N_NODES 8192
#define FDIM    128
#define LOG2E   1.4426950408889634f

#define ROWS_PER_BLOCK 32
#define COLS_PER_BLOCK 1024   // 256 threads * 4 floats

// ---------------------------------------------------------------------------
// Kernel 1: per-node scores via V_WMMA_F32_16X16X4_F32.
//   S = x @ Wmat, Wmat is 128x16 with col0 = wL, col1 = wR, cols 2..15 = 0.
//   Each wave owns a 16-row tile of x, accumulates K=128 in 32 wmma steps.
//   Outputs eL[i] = exp(-S[i,0]),  eR[i] = exp(-(S[i,1] + b)).
// A-fragment (16x4 f32): lane L -> M = L%16, holds K = 2*(L/16)+{0,1}
// B-fragment (4x16 f32): lane L -> N = L%16, holds K = 2*(L/16)+{0,1}
// C/D (16x16 f32): vgpr d, lane L -> M = d + 8*(L/16), N = L%16
// ---------------------------------------------------------------------------
__global__ __launch_bounds__(256) void gat_scores_wmma(
    const float* __restrict__ x, const float* __restrict__ W,
    const float* __restrict__ bptr,
    float* __restrict__ eL, float* __restrict__ eR)
{
    const int lane   = threadIdx.x & 31;
    const int waveId = threadIdx.x >> 5;
    const int tile   = blockIdx.x * 8 + waveId;   // 8 waves per block
    const int rowbase = tile * 16;

    const int m    = lane & 15;    // A row / B col for this lane
    const int half = lane >> 4;

    const float bias = bptr[0];

    // B column source: col 0 -> W[0:128], col 1 -> W[128:256], others zeroed.
    const float* wp  = (m == 0) ? W : (m == 1) ? (W + FDIM) : W;
    const float  bsc = (m < 2) ? 1.0f : 0.0f;

    const float* xrow = x + (size_t)(rowbase + m) * FDIM;

    v8f c = {};
    #pragma unroll 4
    for (int kbase = 0; kbase < FDIM; kbase += 4) {
        const int k0 = kbase + 2 * half;          // even -> 8B aligned
        v2f a = *(const v2f*)(xrow + k0);
        v2f b = *(const v2f*)(wp + k0);
        b.x *= bsc;
        b.y *= bsc;
        // 8 args: (neg_a, A, neg_b, B, c_mod, C, reuse_a, reuse_b)
        c = __builtin_amdgcn_wmma_f32_16x16x4_f32(
                false, a, false, b, (short)0, c, false, false);
    }

    // Lanes with N==0 hold sL for rows rowbase + d + 8*half, N==1 hold sR.
    if (m < 2) {
        float* dst      = (m == 0) ? eL : eR;
        const float add = (m == 0) ? 0.0f : bias;
        #pragma unroll
        for (int d = 0; d < 8; ++d) {
            const int row = rowbase + d + 8 * half;
            const float s = c[d];
            dst[row] = __builtin_amdgcn_exp2f(-(s + add) * LOG2E);
        }
    }
}

// ---------------------------------------------------------------------------
// Kernel 2: pure HBM stream.  out[i,j] = adj[i,j] * 1/(1 + eL[i]*eR[j]).
// Column-stripe register tiling: each block owns a fixed 1024-column stripe
// and walks ROWS_PER_BLOCK rows.  Each thread's 4 eR values live in
// registers for the whole row loop (no LDS, no redundant eR traffic);
// eL[row] is block-uniform -> scalar load from L2.
// adj is read once with non-temporal b128 loads, out written once with
// non-temporal b128 stores, so the 512 MB stream does not thrash L2.
// Minimum per-32B work: 1 NT load, 4x(add+mul, v_rcp_f32, mul), 1 NT store.
// ---------------------------------------------------------------------------
__global__ __launch_bounds__(256) void gat_att_stream(
    const float* __restrict__ adj, const float* __restrict__ eL,
    const float* __restrict__ eR, float* __restrict__ out)
{
    const int tid   = threadIdx.x;
    const int jbase = blockIdx.x * COLS_PER_BLOCK;     // 8 stripes
    const int rbase = blockIdx.y * ROWS_PER_BLOCK;     // 256 row-groups

    const int j = jbase + tid * 4;
    const v4f e = *(const v4f*)(eR + j);   // lives in VGPRs for all rows

    #pragma unroll 4
    for (int r = 0; r < ROWS_PER_BLOCK; ++r) {
        const int row = rbase + r;
        const float t = eL[row];                        // uniform -> s_load
        const size_t idx = (size_t)row * N_NODES + j;

        // keep the stream deep: prefetch the stripe two rows ahead
        if (r + 2 < ROWS_PER_BLOCK)
            __builtin_prefetch(adj + idx + 2u * N_NODES, 0, 0);

        v4f a = __builtin_nontemporal_load((const v4f*)(adj + idx));
        v4f o;
        o.x = a.x * __builtin_amdgcn_rcpf(1.0f + t * e.x);
        o.y = a.y * __builtin_amdgcn_rcpf(1.0f + t * e.y);
        o.z = a.z * __builtin_amdgcn_rcpf(1.0f + t * e.z);
        o.w = a.w * __builtin_amdgcn_rcpf(1.0f + t * e.w);
        __builtin_nontemporal_store(o, (v4f*)(out + idx));
    }
}

// ---------------------------------------------------------------------------
extern "C" void kernel_launch(void* const* d_in, const int* in_sizes, int n_in,
                              void* d_out, int out_size, void* d_ws, size_t ws_size,
                              hipStream_t stream) {
    const float* x   = (const float*)d_in[0];   // (8192, 128)
    const float* adj = (const float*)d_in[1];   // (8192, 8192)
    const float* W   = (const float*)d_in[2];   // (256,)
    const float* b   = (const float*)d_in[3];   // scalar
    float* out = (float*)d_out;

    float* eL = (float*)d_ws;                   // 8192 floats
    float* eR = eL + N_NODES;                   // 8192 floats (64 KB total)

    // 512 row-tiles of 16, 8 waves/block -> 64 blocks
    gat_scores_wmma<<<64, 256, 0, stream>>>(x, W, b, eL, eR);

    // 8 column stripes x 256 row groups = 2048 blocks (8 waves each)
    dim3 grid(N_NODES / COLS_PER_BLOCK, N_NODES / ROWS_PER_BLOCK);
    gat_att_stream<<<grid, 256, 0, stream>>>(adj, eL, eR, out);
}